// PhiLayer_1D_11373073400181
// MI455X (gfx1250) — compile-verified
//
#include <hip/hip_runtime.h>

typedef __attribute__((ext_vector_type(2))) float v2f;
typedef __attribute__((ext_vector_type(4))) float v4f;
typedef __attribute__((ext_vector_type(8))) float v8f;

#define LFEAT 4096   // IN_FEATURES in the reference
#define NKF   8      // K in the reference

// ---------------------------------------------------------------------------
// Stage 1: w[l] = 10 * (2 * sum_k a_k * exp(-(l - nh*floc_k)^2 / (2 b_k^2)) + 1)
// The K=8 reduction is a real (1x8)x(8xL) matmul -> two V_WMMA_F32_16X16X4_F32
// (K=4 each, D->C accumulation). One wave (32 lanes) per 16-column tile.
//
// A (16x4, broadcast rows): lanes 0-15 hold K=0(.x)/K=1(.y), lanes 16-31 hold
// K=2(.x)/K=3(.y)  -> lane supplies a[off+2*hi], a[off+2*hi+1].
// B (4x16): VGPR0 rows K=0(lo)/K=2(hi), VGPR1 rows K=1(lo)/K=3(hi), col = lane&15.
// D: VGPR0 lanes 0-15 = row M=0, N=lane -> acc[0] is the weight for this column.
// ---------------------------------------------------------------------------
__global__ __launch_bounds__(32) void phi_weight_wmma(
    const float* __restrict__ sp, const int* __restrict__ floc,
    const int* __restrict__ nharm, float* __restrict__ w)
{
    const int   tile = blockIdx.x;        // 0 .. LFEAT/16 - 1
    const int   lane = threadIdx.x;       // 0 .. 31 (wave32, EXEC all ones)
    const int   n    = lane & 15;
    const int   hi   = lane >> 4;
    const float lv   = (float)(tile * 16 + n);
    const float nh   = (float)nharm[0];

    v8f acc = {};
#pragma unroll
    for (int off = 0; off < NKF; off += 4) {
        const int   k0 = off + 2 * hi;
        const float a0 = sp[2 * k0 + 0], b0 = sp[2 * k0 + 1];
        const float a1 = sp[2 * k0 + 2], b1 = sp[2 * k0 + 3];
        const float d0 = lv - nh * (float)floc[k0 + 0];
        const float d1 = lv - nh * (float)floc[k0 + 1];
        v2f A, B;
        A.x = a0;
        A.y = a1;
        B.x = __expf(-(d0 * d0) / (2.0f * b0 * b0));
        B.y = __expf(-(d1 * d1) / (2.0f * b1 * b1));
        acc = __builtin_amdgcn_wmma_f32_16x16x4_f32(
            /*neg_a=*/false, A, /*neg_b=*/false, B,
            /*c_mod=*/(short)0, acc, /*reuse_a=*/false, /*reuse_b=*/false);
    }
    if (lane < 16)
        w[tile * 16 + lane] = 10.0f * (2.0f * acc[0] + 1.0f);
}

// ---------------------------------------------------------------------------
// Stage 2: out[i] = w[i mod L] * x[i], HBM-bound streaming multiply.
// float4 accesses -> global_load_b128 / global_store_b128, non-temporal
// (x and out are touched exactly once; only the 16 KB weight table is cached).
// Host guarantees (gridDim*blockDim) % (LFEAT/4) == 0, so each thread's
// weight vec4 is loop-invariant and loaded a single time.
// ---------------------------------------------------------------------------
__global__ __launch_bounds__(256) void phi_scale_stream(
    const float* __restrict__ x, const float* __restrict__ w,
    float* __restrict__ out, unsigned nvec)
{
    const unsigned gtid   = blockIdx.x * blockDim.x + threadIdx.x;
    const unsigned stride = gridDim.x * blockDim.x;  // multiple of LFEAT/4

    const v4f* __restrict__ x4 = (const v4f*)x;
    const v4f* __restrict__ w4 = (const v4f*)w;
    v4f* __restrict__       o4 = (v4f*)out;

    const v4f wv = w4[gtid & (LFEAT / 4 - 1)];

    unsigned i = gtid;
    if (i + 3u * stride < nvec) {
        // 4 independent 128-bit NT load/store pairs (full MLP, no guards)
        v4f v0 = __builtin_nontemporal_load(&x4[i]);
        v4f v1 = __builtin_nontemporal_load(&x4[i + stride]);
        v4f v2 = __builtin_nontemporal_load(&x4[i + 2u * stride]);
        v4f v3 = __builtin_nontemporal_load(&x4[i + 3u * stride]);
        __builtin_nontemporal_store(v0 * wv, &o4[i]);
        __builtin_nontemporal_store(v1 * wv, &o4[i + stride]);
        __builtin_nontemporal_store(v2 * wv, &o4[i + 2u * stride]);
        __builtin_nontemporal_store(v3 * wv, &o4[i + 3u * stride]);
        i += 4u * stride;
    }
    for (; i < nvec; i += stride) {  // tail (empty for the reference shape)
        v4f v = __builtin_nontemporal_load(&x4[i]);
        __builtin_nontemporal_store(v * wv, &o4[i]);
    }
}

extern "C" void kernel_launch(void* const* d_in, const int* in_sizes, int n_in,
                              void* d_out, int out_size, void* d_ws, size_t ws_size,
                              hipStream_t stream) {
    const float* x    = (const float*)d_in[0];  // (128,32,2,4096) f32
    const float* sp   = (const float*)d_in[1];  // (10,160) f32
    const int*   floc = (const int*)d_in[2];    // (8,) i32
    const int*   nh   = (const int*)d_in[3];    // scalar i32
    float*       out  = (float*)d_out;
    float*       w    = (float*)d_ws;           // LFEAT floats = 16 KB scratch

    // Stage 1: 256 one-wave blocks compute the 4096 weights via WMMA.
    phi_weight_wmma<<<LFEAT / 16, 32, 0, stream>>>(sp, floc, nh, w);

    // Stage 2: streaming multiply. stride must be a multiple of LFEAT/4=1024
    // vec4 elements -> blocks multiple of 4 at 256 threads/block.
    const unsigned nvec   = (unsigned)(in_sizes[0] / 4);     // 8,388,608
    unsigned       blocks = (nvec + 1023u) / 1024u;          // ~4 vec4/thread
    blocks = (blocks + 3u) & ~3u;                            // stride % 1024 == 0
    if (blocks == 0) blocks = 4;
    phi_scale_stream<<<blocks, 256, 0, stream>>>(x, w, out, nvec);
}